// NMS_76656576299736
// MI455X (gfx1250) — compile-verified
//
#include <hip/hip_runtime.h>
#include <stdint.h>

#define N_BOXES   8192
#define WORDS     (N_BOXES / 64)      // 128 u64 words per mask row
#define SCORE_THR 0.5f
#define IOU_THR   0.5f

typedef unsigned int v4u __attribute__((ext_vector_type(4)));
typedef int          v4i __attribute__((ext_vector_type(4)));
typedef int          v8i __attribute__((ext_vector_type(8)));

#if defined(__gfx1250__) && __has_builtin(__builtin_amdgcn_tensor_load_to_lds) && \
    __has_builtin(__builtin_amdgcn_tensor_store_from_lds) && \
    __has_builtin(__builtin_amdgcn_s_wait_tensorcnt)
#define USE_TDM 1
#else
#define USE_TDM 0
#endif

#if USE_TDM
// ---- Tensor DMA descriptor builders (CDNA5 ISA ch.8 layout) ----------------
__device__ __forceinline__ v4u tdm_g0(uint32_t lds_off, uint64_t gaddr) {
    v4u g;
    g[0] = 1u;                                   // count=1, is_restore=0
    g[1] = lds_off;                              // lds_addr [63:32]
    g[2] = (uint32_t)gaddr;                      // global_addr [95:64]
    g[3] = (uint32_t)((gaddr >> 32) & 0x1FFFFFFu) | (2u << 30); // addr[56:32] | type=2
    return g;
}
// data_size fixed at 8 bytes (code 3). Fields per D# group-1 bit layout.
__device__ __forceinline__ v8i tdm_g1(uint32_t tensor_d0, uint32_t tensor_d1,
                                      uint32_t tile_d0, uint32_t tile_d1,
                                      uint32_t tile_d2, uint64_t stride_d0) {
    v8i g;
    g[0] = (int)(3u << 16);                                            // data_size=8B
    g[1] = (int)((tensor_d0 & 0xFFFFu) << 16);                         // tensor_dim0 lo
    g[2] = (int)(((tensor_d0 >> 16) & 0xFFFFu) | ((tensor_d1 & 0xFFFFu) << 16));
    g[3] = (int)(((tensor_d1 >> 16) & 0xFFFFu) | ((tile_d0 & 0xFFFFu) << 16));
    g[4] = (int)((tile_d1 & 0xFFFFu) | ((tile_d2 & 0xFFFFu) << 16));
    g[5] = (int)(stride_d0 & 0xFFFFFFFFull);                           // dim0 stride lo
    g[6] = (int)((stride_d0 >> 32) & 0xFFFFull);                       // stride hi | stride1=0
    g[7] = 0;
    return g;
}
__device__ __forceinline__ void tdm_load(v4u g0, v8i g1) {
    v4i z4 = {0, 0, 0, 0};
#if __clang_major__ >= 23
    v8i z8 = {0, 0, 0, 0, 0, 0, 0, 0};
    __builtin_amdgcn_tensor_load_to_lds(g0, g1, z4, z4, z8, 0);
#else
    __builtin_amdgcn_tensor_load_to_lds(g0, g1, z4, z4, 0);
#endif
}
__device__ __forceinline__ void tdm_store(v4u g0, v8i g1) {
    v4i z4 = {0, 0, 0, 0};
#if __clang_major__ >= 23
    v8i z8 = {0, 0, 0, 0, 0, 0, 0, 0};
    __builtin_amdgcn_tensor_store_from_lds(g0, g1, z4, z4, z8, 0);
#else
    __builtin_amdgcn_tensor_store_from_lds(g0, g1, z4, z4, 0);
#endif
}
__device__ __forceinline__ uint32_t lds_off_of(const void* p) {
    return (uint32_t)(uintptr_t)p;   // flat LDS pointer: low 32 bits = LDS offset
}
#endif

// ---------------------------------------------------------------------------
// Kernel A: stable descending sort by score (bitonic, single block, dyn LDS)
// key = (~ord(score) << 32) | original_index  -> ascending sort == stable
// argsort(-score). Invalid (score < 0.5) mapped to -inf -> sorts last.
// ---------------------------------------------------------------------------
extern "C" __global__ __launch_bounds__(1024)
void nms_sort_kernel(const float* __restrict__ xyxy,
                     const float* __restrict__ conf,
                     float* __restrict__ boxes_s,
                     float* __restrict__ scores_s,
                     int* __restrict__ valid_s) {
    extern __shared__ unsigned long long skey[];   // 8192 * 8B = 64 KB
    const int tid = threadIdx.x;

    for (int i = tid; i < N_BOXES; i += 1024) {
        float s   = conf[i];
        float key = (s >= SCORE_THR) ? s : -__builtin_inff();
        uint32_t u   = __float_as_uint(key);
        uint32_t ord = (u & 0x80000000u) ? ~u : (u | 0x80000000u);
        skey[i] = ((unsigned long long)(~ord) << 32) | (uint32_t)i;
    }
    __syncthreads();

    for (int k = 2; k <= N_BOXES; k <<= 1) {
        for (int j = k >> 1; j > 0; j >>= 1) {
            for (int i = tid; i < N_BOXES; i += 1024) {
                int ij = i ^ j;
                if (ij > i) {
                    unsigned long long a = skey[i];
                    unsigned long long b = skey[ij];
                    bool up = ((i & k) == 0);
                    if ((a > b) == up) { skey[i] = b; skey[ij] = a; }
                }
            }
            __syncthreads();
        }
    }

    for (int i = tid; i < N_BOXES; i += 1024) {
        int src = (int)(skey[i] & 0xFFFFFFFFull);
        float4 b = ((const float4*)xyxy)[src];
        ((float4*)boxes_s)[i] = b;
        float sc = conf[src];
        scores_s[i] = sc;
        valid_s[i]  = (sc >= SCORE_THR) ? 1 : 0;
    }
}

// ---------------------------------------------------------------------------
// Kernel B: suppression bitmask. Block = 64 rows x 256 cols tile.
// thread (r, w): row i0+r, 64-bit word w of the column tile.
// Row boxes + column boxes staged to LDS and the 2 KB mask tile written back
// via the CDNA5 Tensor Data Mover (TENSORcnt path).
// ---------------------------------------------------------------------------
extern "C" __global__ __launch_bounds__(256)
void nms_mask_kernel(const float* __restrict__ boxes_s,
                     unsigned long long* __restrict__ mask) {
    const int i0 = blockIdx.x * 64;
    const int j0 = blockIdx.y * 256;
    const int r  = threadIdx.x >> 2;   // 0..63  row in tile
    const int w  = threadIdx.x & 3;    // 0..3   word in tile
    const int i  = i0 + r;

    // Whole tile strictly at-or-below the diagonal -> all words zero.
    // (Uniform per block: no divergent barrier.)
    if (j0 + 255 <= i0) {
        mask[(size_t)i * WORDS + (j0 >> 6) + w] = 0ull;
        return;
    }

    __shared__ float tileCols[256 * 4];            // 4 KB column boxes
    __shared__ float tileRows[64 * 4];             // 1 KB row boxes
    __shared__ unsigned long long wtile[64 * 4];   // 2 KB mask tile (row-major)

#if USE_TDM
    if ((threadIdx.x >> 5) == 0) {   // wave 0 issues the DMA descriptors
        tdm_load(tdm_g0(lds_off_of(tileCols),
                        (uint64_t)(uintptr_t)(boxes_s + (size_t)j0 * 4)),
                 tdm_g1(1u << 20, 1u, 512u, 0u, 0u, 512u));   // 4 KB, 1-D
        tdm_load(tdm_g0(lds_off_of(tileRows),
                        (uint64_t)(uintptr_t)(boxes_s + (size_t)i0 * 4)),
                 tdm_g1(1u << 20, 1u, 128u, 0u, 0u, 128u));   // 1 KB, 1-D
        __builtin_amdgcn_s_wait_tensorcnt(0);
    }
    __syncthreads();
#else
    ((float4*)tileCols)[threadIdx.x] = ((const float4*)boxes_s)[j0 + threadIdx.x];
    if (threadIdx.x < 64)
        ((float4*)tileRows)[threadIdx.x] = ((const float4*)boxes_s)[i0 + threadIdx.x];
    __syncthreads();
#endif

    float ax1 = tileRows[r * 4 + 0];
    float ay1 = tileRows[r * 4 + 1];
    float ax2 = tileRows[r * 4 + 2];
    float ay2 = tileRows[r * 4 + 3];
    float areaA = (ax2 - ax1) * (ay2 - ay1);

    unsigned long long word = 0ull;
    const int jbase = j0 + w * 64;
    for (int c = 0; c < 64; ++c) {
        int lc = w * 64 + c;
        float bx1 = tileCols[lc * 4 + 0];
        float by1 = tileCols[lc * 4 + 1];
        float bx2 = tileCols[lc * 4 + 2];
        float by2 = tileCols[lc * 4 + 3];
        float ix1 = fmaxf(ax1, bx1);
        float iy1 = fmaxf(ay1, by1);
        float ix2 = fminf(ax2, bx2);
        float iy2 = fminf(ay2, by2);
        float iw  = fmaxf(ix2 - ix1, 0.0f);
        float ih  = fmaxf(iy2 - iy1, 0.0f);
        float inter = iw * ih;
        float areaB = (bx2 - bx1) * (by2 - by1);
        float uni   = areaA + areaB - inter;
        int j = jbase + c;
        bool sup = (inter > IOU_THR * fmaxf(uni, 1e-9f)) && (j > i);
        word |= ((unsigned long long)(sup ? 1u : 0u)) << c;
    }

#if USE_TDM
    wtile[r * 4 + w] = word;
    __syncthreads();                 // full tile in LDS before DMA reads it
    if ((threadIdx.x >> 5) == 0) {
        // 2-D store: 4 u64 words per row, 64 rows, global row stride = WORDS u64
        tdm_store(tdm_g0(lds_off_of(wtile),
                         (uint64_t)(uintptr_t)(mask + (size_t)i0 * WORDS + (j0 >> 6))),
                  tdm_g1(1u << 20, 1u << 20, 4u, 64u, 0u, (uint64_t)WORDS));
        // S_ENDPGM drains TENSORcnt for the issuing wave; LDS lives until the
        // workgroup (i.e. this wave) completes.
    }
#else
    mask[(size_t)i * WORDS + (j0 >> 6) + w] = word;
#endif
}

// ---------------------------------------------------------------------------
// Kernel C: sequential greedy scan over mask rows (one block, remv in LDS).
// ---------------------------------------------------------------------------
extern "C" __global__ __launch_bounds__(128)
void nms_scan_kernel(const unsigned long long* __restrict__ mask,
                     const int* __restrict__ valid_s,
                     int* __restrict__ keep) {
    __shared__ unsigned long long remv[WORDS];
    __shared__ int flag;
    const int t = threadIdx.x;
    remv[t] = 0ull;
    __syncthreads();

    for (int i = 0; i < N_BOXES; ++i) {
        if (i + 1 < N_BOXES)
            __builtin_prefetch(&mask[(size_t)(i + 1) * WORDS + t], 0, 0);
        if (t == 0) {
            int k = valid_s[i] && !((remv[i >> 6] >> (i & 63)) & 1ull);
            flag = k;
            keep[i] = k;
        }
        __syncthreads();
        if (flag)
            remv[t] |= mask[(size_t)i * WORDS + t];
        __syncthreads();
    }
}

// ---------------------------------------------------------------------------
// Kernel D: masked output. d_out = [boxes_s|scores_s masked (N,5)] ++ [keep (N,)]
// ---------------------------------------------------------------------------
extern "C" __global__ __launch_bounds__(256)
void nms_out_kernel(const float* __restrict__ boxes_s,
                    const float* __restrict__ scores_s,
                    const int* __restrict__ keep,
                    float* __restrict__ out) {
    int i = blockIdx.x * blockDim.x + threadIdx.x;
    if (i >= N_BOXES) return;
    bool k = keep[i] != 0;
    float4 b = ((const float4*)boxes_s)[i];
    out[i * 5 + 0] = k ? b.x : 0.0f;
    out[i * 5 + 1] = k ? b.y : 0.0f;
    out[i * 5 + 2] = k ? b.z : 0.0f;
    out[i * 5 + 3] = k ? b.w : 0.0f;
    out[i * 5 + 4] = k ? scores_s[i] : 0.0f;
    out[N_BOXES * 5 + i] = k ? 1.0f : 0.0f;
}

// ---------------------------------------------------------------------------
extern "C" void kernel_launch(void* const* d_in, const int* in_sizes, int n_in,
                              void* d_out, int out_size, void* d_ws, size_t ws_size,
                              hipStream_t stream) {
    const float* xyxy = (const float*)d_in[0];   // (N,4) f32
    const float* conf = (const float*)d_in[1];   // (N,)  f32

    char* ws = (char*)d_ws;
    float* boxes_s  = (float*)(ws + 0);                                   // 128 KB
    float* scores_s = (float*)(ws + (size_t)N_BOXES * 4 * sizeof(float)); // +32 KB
    int*   valid_s  = (int*)  (ws + (size_t)N_BOXES * 5 * sizeof(float)); // +32 KB
    int*   keep     = (int*)  (ws + (size_t)N_BOXES * 6 * sizeof(float)); // +32 KB
    unsigned long long* mask =
        (unsigned long long*)(ws + 256 * 1024);                           // 8 MB

    nms_sort_kernel<<<1, 1024, 64 * 1024, stream>>>(xyxy, conf,
                                                    boxes_s, scores_s, valid_s);

    dim3 gridB(N_BOXES / 64, N_BOXES / 256);   // 128 x 32 = 4096 blocks
    nms_mask_kernel<<<gridB, 256, 0, stream>>>(boxes_s, mask);

    nms_scan_kernel<<<1, WORDS, 0, stream>>>(mask, valid_s, keep);

    nms_out_kernel<<<(N_BOXES + 255) / 256, 256, 0, stream>>>(
        boxes_s, scores_s, keep, (float*)d_out);
}